// MambaBlock_14302241096144
// MI455X (gfx1250) — compile-verified
//
#include <hip/hip_runtime.h>
#include <hip/hip_bf16.h>
#include <math.h>

typedef __attribute__((ext_vector_type(2))) float v2f;
typedef __attribute__((ext_vector_type(8))) float v8f;
typedef __attribute__((ext_vector_type(4))) unsigned int v4u;
typedef __attribute__((ext_vector_type(8))) int v8i;
typedef __attribute__((ext_vector_type(4))) int v4i;

#define L_SEQ  2048
#define DMODEL 1024
#define DINNER 2048
#define DSTATE 16

__device__ __forceinline__ float silu_(float x) { return x / (1.f + __expf(-x)); }

// ---------------------------------------------------------------------------
// Fragment helpers: chunk = 16 k-values = 4 wmma steps.
// A-fragment lane map: lane holds A[m = lane%16][k = k0 + 2*(lane>>4) + v]
// B-fragment lane map: lane holds W[n = lane%16][k = k0 + 2*(lane>>4) + v]
// ---------------------------------------------------------------------------
template<int NT>
__device__ __forceinline__ void load_chunk(v2f (&aF)[2][4], v2f (&bF)[NT][4],
                                           const float* __restrict__ Arow0,
                                           const float* __restrict__ Arow1,
                                           const float* __restrict__ Wrow,
                                           int K, int k0)
{
    #pragma unroll
    for (int s = 0; s < 4; ++s) {
        aF[0][s] = *(const v2f*)(Arow0 + k0 + 4 * s);
        aF[1][s] = *(const v2f*)(Arow1 + k0 + 4 * s);
        #pragma unroll
        for (int nt = 0; nt < NT; ++nt)
            bF[nt][s] = *(const v2f*)(Wrow + (size_t)nt * 16 * K + k0 + 4 * s);
    }
}

template<int NT>
__device__ __forceinline__ void compute_chunk(v8f (&acc)[2][NT],
                                              const v2f (&aF)[2][4],
                                              const v2f (&bF)[NT][4])
{
    #pragma unroll
    for (int s = 0; s < 4; ++s)
        #pragma unroll
        for (int mt = 0; mt < 2; ++mt)
            #pragma unroll
            for (int nt = 0; nt < NT; ++nt)
                acc[mt][nt] = __builtin_amdgcn_wmma_f32_16x16x4_f32(
                    false, aF[mt][s], false, bF[nt][s],
                    (short)0, acc[mt][nt], false, false);
}

// ---------------------------------------------------------------------------
// C[M,N] = A[M,K] (row stride lda) @ W[N,K]^T, fp32 WMMA 16x16x4.
// Wave tile 32 x (16*NT); K pipelined in 16-wide chunks, register double buffer.
// ---------------------------------------------------------------------------
template<int NT, bool SOFTPLUS>
__global__ __launch_bounds__(256, 1)
void wmma_gemm_nt(const float* __restrict__ A, int lda,
                  const float* __restrict__ W,
                  const float* __restrict__ bias,
                  float* __restrict__ C, int ldc,
                  int M, int N, int K)
{
    const int lane = threadIdx.x & 31;
    const int wid  = (blockIdx.x * blockDim.x + threadIdx.x) >> 5;
    const int wavesM = M >> 5;
    const int totalWaves = wavesM * (N / (16 * NT));
    if (wid >= totalWaves) return;                   // whole-wave uniform exit

    const int m0   = (wid % wavesM) * 32;
    const int n0   = (wid / wavesM) * (16 * NT);
    const int half = lane >> 4;
    const int lm   = lane & 15;

    const float* Arow0 = A + (size_t)(m0 + lm) * lda + 2 * half;
    const float* Arow1 = Arow0 + (size_t)16 * lda;
    const float* Wrow  = W + (size_t)(n0 + lm) * K + 2 * half;

    v8f acc[2][NT] = {};
    v2f a0[2][4], b0[NT][4];
    v2f a1[2][4], b1[NT][4];

    load_chunk<NT>(a0, b0, Arow0, Arow1, Wrow, K, 0);
    int k0 = 0;
    for (; k0 + 32 <= K; k0 += 32) {
        load_chunk<NT>(a1, b1, Arow0, Arow1, Wrow, K, k0 + 16);
        compute_chunk<NT>(acc, a0, b0);
        if (k0 + 32 < K)
            load_chunk<NT>(a0, b0, Arow0, Arow1, Wrow, K, k0 + 32);
        compute_chunk<NT>(acc, a1, b1);
    }
    if (k0 < K)                                      // odd trailing chunk (e.g. K=16)
        compute_chunk<NT>(acc, a0, b0);

    #pragma unroll
    for (int mt = 0; mt < 2; ++mt) {
        #pragma unroll
        for (int nt = 0; nt < NT; ++nt) {
            const int n = n0 + nt * 16 + lm;
            float bv = 0.f;
            if (SOFTPLUS) bv = bias[n];
            #pragma unroll
            for (int r = 0; r < 8; ++r) {
                const int m = m0 + mt * 16 + r + 8 * half;
                float v = acc[mt][nt][r];
                if (SOFTPLUS) {
                    v += bv;
                    v = (v > 20.f) ? v : logf(1.f + __expf(v));
                }
                C[(size_t)m * ldc + n] = v;
            }
        }
    }
}

// ---------------------------------------------------------------------------
// Depthwise causal conv1d (k=4) + bias + SiLU.  xz is [L, 4096]; xc = cols 0..2047.
// ---------------------------------------------------------------------------
__global__ void conv_silu_kernel(const float* __restrict__ xz,
                                 const float* __restrict__ conv_w,
                                 const float* __restrict__ conv_b,
                                 float* __restrict__ xc_out)
{
    int idx = blockIdx.x * blockDim.x + threadIdx.x;   // over L * DINNER, d fastest
    int l = idx / DINNER;
    int d = idx % DINNER;
    float s = conv_b[d];
    #pragma unroll
    for (int j = 0; j < 4; ++j) {
        int t = l - 3 + j;
        float xv = (t >= 0) ? xz[(size_t)t * (2 * DINNER) + d] : 0.f;
        s = fmaf(conv_w[d * 4 + j], xv, s);
    }
    xc_out[idx] = silu_(s);
}

// ---------------------------------------------------------------------------
// Selective scan. 4 lanes per channel d, 4 states each; h kept in registers.
// B/C rows staged into LDS via the Tensor Data Mover:
//   tile = 128 rows x 32 floats (x_dbl cols 16..47), row stride 48 floats.
// Wave 0 issues TENSOR_LOAD_TO_LDS, waits TENSORcnt, barrier publishes tile.
// Fused:  y = (scan + u*D[d]) * silu(res).
// ---------------------------------------------------------------------------
__global__ void scan_kernel(const float* __restrict__ delta,
                            const float* __restrict__ u,      // xc after conv+silu
                            const float* __restrict__ xdbl,   // [L, 48]
                            const float* __restrict__ A_log,  // [DINNER, 16]
                            const float* __restrict__ Dv,     // [DINNER]
                            const float* __restrict__ xz,     // res = cols 2048..4095
                            float* __restrict__ y_out)        // [L, DINNER]
{
    const int t   = threadIdx.x;                // 256 threads: 64 channels x 4 lanes
    const int d   = blockIdx.x * 64 + (t >> 2);
    const int sub = t & 3;                      // state group: n = sub*4 .. sub*4+3

    float A[4], h[4] = {0.f, 0.f, 0.f, 0.f};
    #pragma unroll
    for (int i = 0; i < 4; ++i)
        A[i] = -__expf(A_log[d * DSTATE + sub * 4 + i]);
    const float Dd = Dv[d];

    __shared__ float sBC[128 * 32];             // [row][0..15]=B, [16..31]=C

    const unsigned ldsBase = (unsigned)(uintptr_t)(&sBC[0]);  // LDS byte offset

    for (int lc = 0; lc < L_SEQ; lc += 128) {
        if (t < 32) {                           // wave 0 drives the TDM
            unsigned long long ga =
                (unsigned long long)(uintptr_t)(xdbl + (size_t)lc * 48 + 16);
            // D# group 0: count=1 | lds_addr | global_addr | type=2
            v4u g0 = { 1u,
                       ldsBase,
                       (unsigned)(ga & 0xFFFFFFFFu),
                       (unsigned)((ga >> 32) & 0x01FFFFFFu) | 0x80000000u };
            // D# group 1: data_size=4B; tensor 32 x 128 (stride 48); tile 32 x 128
            v8i g1 = { (int)(2u << 16),          // wg_mask=0, data_size=2 (4B)
                       (int)(32u << 16),         // tensor_dim0 = 32
                       (int)(128u << 16),        // tensor_dim1 = 128
                       (int)(32u << 16),         // tile_dim0   = 32
                       (int)128,                 // tile_dim1   = 128, tile_dim2=0
                       (int)48,                  // tensor_dim0_stride = 48
                       0, 0 };
            v4i g2 = { 0, 0, 0, 0 };
            v4i g3 = { 0, 0, 0, 0 };
            v8i g4 = { 0, 0, 0, 0, 0, 0, 0, 0 };
            __builtin_amdgcn_tensor_load_to_lds(g0, g1, g2, g3, g4, 0);
            __builtin_amdgcn_s_wait_tensorcnt(0);
        }
        __syncthreads();                        // tile visible to all waves

        for (int li = 0; li < 128; ++li) {
            const int l  = lc + li;
            const float dl = delta[(size_t)l * DINNER + d];
            const float uv = u[(size_t)l * DINNER + d];
            float y = 0.f;
            #pragma unroll
            for (int i = 0; i < 4; ++i) {
                const int n = sub * 4 + i;
                float dA = __expf(dl * A[i]);
                h[i] = fmaf(dA, h[i], dl * sBC[li * 32 + n] * uv);
                y = fmaf(h[i], sBC[li * 32 + 16 + n], y);
            }
            // reduce y over the 4-lane state group
            y += __shfl_xor(y, 1, 32);
            y += __shfl_xor(y, 2, 32);
            if (sub == 0) {
                float yf = fmaf(uv, Dd, y);
                float r  = xz[(size_t)l * (2 * DINNER) + DINNER + d];
                y_out[(size_t)l * DINNER + d] = yf * silu_(r);
            }
        }
        __syncthreads();                        // all readers done before next DMA
    }
}

// ---------------------------------------------------------------------------
extern "C" void kernel_launch(void* const* d_in, const int* in_sizes, int n_in,
                              void* d_out, int out_size, void* d_ws, size_t ws_size,
                              hipStream_t stream)
{
    const float* x      = (const float*)d_in[0];  // [2048, 1024]
    const float* W_in   = (const float*)d_in[1];  // [4096, 1024]
    const float* conv_w = (const float*)d_in[2];  // [2048, 1, 4]
    const float* conv_b = (const float*)d_in[3];  // [2048]
    const float* W_x    = (const float*)d_in[4];  // [48, 2048]
    const float* W_dt   = (const float*)d_in[5];  // [2048, 16]
    const float* b_dt   = (const float*)d_in[6];  // [2048]
    const float* A_log  = (const float*)d_in[7];  // [2048, 16]
    const float* Dv     = (const float*)d_in[8];  // [2048]
    const float* W_out  = (const float*)d_in[9];  // [1024, 2048]
    float* out = (float*)d_out;                   // [2048, 1024]

    float* ws    = (float*)d_ws;
    float* xz    = ws;                              // L * 4096
    float* xc    = xz    + (size_t)L_SEQ * 4096;    // L * 2048
    float* xdbl  = xc    + (size_t)L_SEQ * 2048;    // L * 48
    float* delta = xdbl  + (size_t)L_SEQ * 48;      // L * 2048
    float* yv    = delta + (size_t)L_SEQ * 2048;    // L * 2048

    // 1) in_proj: xz = x @ W_in^T   (M=2048, N=4096, K=1024); wave tile 32x64
    {
        int waves = (L_SEQ / 32) * (4096 / 64);     // 4096 waves
        wmma_gemm_nt<4, false><<<waves / 8, 256, 0, stream>>>(
            x, DMODEL, W_in, nullptr, xz, 4096, L_SEQ, 4096, DMODEL);
    }
    // 2) depthwise causal conv + SiLU
    conv_silu_kernel<<<(L_SEQ * DINNER) / 256, 256, 0, stream>>>(xz, conv_w, conv_b, xc);

    // 3) x_dbl = xc @ W_x^T   (M=2048, N=48, K=2048); wave tile 32x48
    {
        int waves = (L_SEQ / 32);                   // 64 waves, NT=3
        wmma_gemm_nt<3, false><<<waves / 8, 256, 0, stream>>>(
            xc, DINNER, W_x, nullptr, xdbl, 48, L_SEQ, 48, DINNER);
    }
    // 4) delta = softplus(x_dbl[:, :16] @ W_dt^T + b_dt)  (M=2048, N=2048, K=16, lda=48)
    {
        int waves = (L_SEQ / 32) * (DINNER / 64);   // 2048 waves
        wmma_gemm_nt<4, true><<<waves / 8, 256, 0, stream>>>(
            xdbl, 48, W_dt, b_dt, delta, DINNER, L_SEQ, DINNER, 16);
    }
    // 5) selective scan + skip + gate (TDM-staged B/C)
    scan_kernel<<<DINNER / 64, 256, 0, stream>>>(delta, xc, xdbl, A_log, Dv, xz, yv);

    // 6) out_proj: out = yv @ W_out^T  (M=2048, N=1024, K=2048); wave tile 32x64
    {
        int waves = (L_SEQ / 32) * (DMODEL / 64);   // 1024 waves
        wmma_gemm_nt<4, false><<<waves / 8, 256, 0, stream>>>(
            yv, DINNER, W_out, nullptr, out, DMODEL, L_SEQ, DMODEL, DINNER);
    }
}